// GraphAttentionLayer_13091060318912
// MI455X (gfx1250) — compile-verified
//
#include <hip/hip_runtime.h>
#include <math.h>

// GAT layer: n=4096 nodes, in_f=256, heads=4, hidden=64.
// Inputs: d_in[0]=h (f32 4096x256), d_in[1]=adj (bool -> uint8, 4096x4096),
//         d_in[2]=W (f32 256x256, stored [out_f][in_f]), d_in[3]=attn_w (f32 128).
// Output: f32 4096x256.

#define N_NODES 4096
#define IN_F    256
#define HEADS   4
#define HIDDEN  64
#define OUT_F   256          // HEADS*HIDDEN
#define NEG_SLOPE 0.2f
#define PSTR    66           // padded LDS row stride (even: 8B-aligned b64; %64==2: no bank conflicts)

typedef __attribute__((ext_vector_type(2))) float v2f;
typedef __attribute__((ext_vector_type(8))) float v8f;

// V_WMMA_F32_16X16X4_F32: D(16x16 f32) = A(16x4 f32) x B(4x16 f32) + C
// A layout (wave32): lane m=(l&15): VGPR v holds K = 2*(l>>4)+v
// B layout symmetric: lane n=(l&15): VGPR v holds K = 2*(l>>4)+v
// C/D layout: VGPR v, lane l: M = v + 8*(l>>4), N = l&15
__device__ __forceinline__ v8f wmma_f32_k4(v2f a, v2f b, v8f c) {
  return __builtin_amdgcn_wmma_f32_16x16x4_f32(
      /*neg_a=*/false, a, /*neg_b=*/false, b,
      /*c_mod=*/(short)0, c, /*reuse_a=*/false, /*reuse_b=*/false);
}

// ---------------------------------------------------------------------------
// Kernel 1: g = h @ W^T   (4096x256) = (4096x256)(256x256)
// grid = (N/16, OUT_F/64), block = 128 (4 waves; wave w owns N-tile w)
// ---------------------------------------------------------------------------
__global__ void gemm_g_kernel(const float* __restrict__ h,
                              const float* __restrict__ W,
                              float* __restrict__ g) {
  const int lane  = threadIdx.x & 31;
  const int wave  = threadIdx.x >> 5;
  const int m0    = blockIdx.x * 16;
  const int n0    = blockIdx.y * 64 + wave * 16;
  const int idx16 = lane & 15;
  const int khalf = (lane >> 4) * 2;     // 0 or 2

  const float* arow = h + (size_t)(m0 + idx16) * IN_F + khalf;
  const float* brow = W + (size_t)(n0 + idx16) * IN_F + khalf;  // W is [N][K]

  v8f acc = {};
#pragma unroll 4
  for (int k = 0; k < IN_F; k += 4) {
    v2f a = *(const v2f*)(arow + k);     // global_load_b64
    v2f b = *(const v2f*)(brow + k);     // global_load_b64
    acc = wmma_f32_k4(a, b, acc);
  }

  const int row = m0 + (lane >> 4) * 8;
  float* out = g + (size_t)row * OUT_F + n0 + idx16;
#pragma unroll
  for (int v = 0; v < 8; ++v) out[(size_t)v * OUT_F] = acc[v];
}

// ---------------------------------------------------------------------------
// Kernel 2: score_l[i,h] = dot(g[i,h,:], a_l), score_r likewise.
// One thread per (node, head). attn_w staged in LDS.
// ---------------------------------------------------------------------------
__global__ void scores_kernel(const float* __restrict__ g,
                              const float* __restrict__ attn_w,
                              float* __restrict__ sl,
                              float* __restrict__ sr) {
  __shared__ float aw[2 * HIDDEN];
  if (threadIdx.x < 2 * HIDDEN) aw[threadIdx.x] = attn_w[threadIdx.x];
  __syncthreads();

  const int t    = blockIdx.x * blockDim.x + threadIdx.x;  // i*HEADS + head
  const int i    = t >> 2;
  const int head = t & 3;
  const float* grow = g + (size_t)i * OUT_F + head * HIDDEN;

  float accl = 0.f, accr = 0.f;
#pragma unroll 4
  for (int f = 0; f < HIDDEN; ++f) {
    float v = grow[f];
    accl += v * aw[f];
    accr += v * aw[HIDDEN + f];
  }
  sl[t] = accl;
  sr[t] = accr;
}

// ---------------------------------------------------------------------------
// Kernel 3: per-row softmax stats. One 256-thread block per node i.
// Online (max, sum) per head, then LDS tree reduce. Writes row max M and 1/Z.
// ---------------------------------------------------------------------------
__global__ void softmax_stats_kernel(const float* __restrict__ sl,
                                     const float* __restrict__ sr,
                                     const unsigned char* __restrict__ adj,
                                     float* __restrict__ Mout,
                                     float* __restrict__ IZout) {
  const int i = blockIdx.x;
  const int t = threadIdx.x;

  float sli[HEADS];
#pragma unroll
  for (int hh = 0; hh < HEADS; ++hh) sli[hh] = sl[i * HEADS + hh];

  float m[HEADS], s[HEADS];
#pragma unroll
  for (int hh = 0; hh < HEADS; ++hh) { m[hh] = -INFINITY; s[hh] = 0.f; }

  const unsigned char* arow = adj + (size_t)i * N_NODES;
  for (int j = t; j < N_NODES; j += 256) {
    if (j + 1024 < N_NODES) __builtin_prefetch(arow + j + 1024, 0, 1);  // global_prefetch_b8
    if (arow[j]) {
#pragma unroll
      for (int hh = 0; hh < HEADS; ++hh) {
        float e = sli[hh] + sr[j * HEADS + hh];
        e = (e >= 0.f) ? e : NEG_SLOPE * e;
        if (e > m[hh]) {
          s[hh] = s[hh] * __expf(m[hh] - e) + 1.f;  // exp(-inf)=0 on first hit
          m[hh] = e;
        } else {
          s[hh] += __expf(e - m[hh]);
        }
      }
    }
  }

  __shared__ float smax[HEADS][256];
  __shared__ float ssum[HEADS][256];
#pragma unroll
  for (int hh = 0; hh < HEADS; ++hh) { smax[hh][t] = m[hh]; ssum[hh][t] = s[hh]; }
  __syncthreads();

  for (int off = 128; off > 0; off >>= 1) {
    if (t < off) {
#pragma unroll
      for (int hh = 0; hh < HEADS; ++hh) {
        float m1 = smax[hh][t],        s1 = ssum[hh][t];
        float m2 = smax[hh][t + off],  s2 = ssum[hh][t + off];
        if (m2 > m1) { float tm = m1; m1 = m2; m2 = tm;
                       float ts = s1; s1 = s2; s2 = ts; }
        if (m2 != -INFINITY) s1 += s2 * __expf(m2 - m1);
        smax[hh][t] = m1; ssum[hh][t] = s1;
      }
    }
    __syncthreads();
  }
  if (t < HEADS) {
    Mout[i * HEADS + t]  = smax[t][0];
    IZout[i * HEADS + t] = 1.0f / ssum[t][0];  // diagonal guarantees >=1 member
  }
}

// ---------------------------------------------------------------------------
// Kernel 4: out[i, h*64+f] = sum_j P[i,j,h] * g[j, h*64+f]
// Block = (16 i-rows, one head), 4 waves; wave w owns f-slice [w*16, w*16+16).
// Per 64-j block: cooperatively build the 16x64 probability tile AND stage the
// 64x64 g block transposed in LDS, then 16 f32 WMMAs per wave fed entirely by
// conflict-free ds_load_b64 (padded stride 66).
// grid = (N/16, HEADS), block = 128.
// ---------------------------------------------------------------------------
__global__ void attn_gemm_kernel(const float* __restrict__ g,
                                 const float* __restrict__ sl,
                                 const float* __restrict__ sr,
                                 const float* __restrict__ Mst,
                                 const float* __restrict__ IZ,
                                 const unsigned char* __restrict__ adj,
                                 float* __restrict__ out) {
  const int i0   = blockIdx.x * 16;
  const int head = blockIdx.y;
  const int lane = threadIdx.x & 31;
  const int wave = threadIdx.x >> 5;

  __shared__ float sP [16 * PSTR];         // probability tile  [m][jj]
  __shared__ float sGT[64 * PSTR];         // g block transposed [f][jj]
  __shared__ float sSL[16], sM[16], sIZ[16];

  if (threadIdx.x < 16) {
    const int r = threadIdx.x;
    sSL[r] = sl [(i0 + r) * HEADS + head];
    sM [r] = Mst[(i0 + r) * HEADS + head];
    sIZ[r] = IZ [(i0 + r) * HEADS + head];
  }
  __syncthreads();

  const int idx16 = lane & 15;
  const int khalf = (lane >> 4) * 2;       // 0 or 2
  const float* gblk = g + (size_t)head * HIDDEN;

  v8f acc = {};
  for (int j0 = 0; j0 < N_NODES; j0 += 64) {
    // Phase 1a: probability tile (16 rows x 64 j), 8 elements per thread.
#pragma unroll
    for (int r = 0; r < 8; ++r) {
      const int e    = threadIdx.x + r * 128;
      const int mrow = e >> 6;
      const int jj   = e & 63;
      const int j    = j0 + jj;
      if (j + 64 < N_NODES)                 // prefetch next adjacency block
        __builtin_prefetch(adj + (size_t)(i0 + mrow) * N_NODES + j + 64, 0, 1);
      float ev = sSL[mrow] + sr[j * HEADS + head];
      ev = (ev >= 0.f) ? ev : NEG_SLOPE * ev;
      float p = adj[(size_t)(i0 + mrow) * N_NODES + j]
                    ? (__expf(ev - sM[mrow]) * sIZ[mrow]) : 0.f;
      sP[mrow * PSTR + jj] = p;
    }
    // Phase 1b: stage g[j0..j0+63][head*64..+63] transposed (coalesced reads).
#pragma unroll 8
    for (int r = 0; r < 32; ++r) {
      const int e  = threadIdx.x + r * 128;
      const int jj = e >> 6;
      const int ff = e & 63;
      sGT[ff * PSTR + jj] = gblk[(size_t)(j0 + jj) * OUT_F + ff];
    }
    __syncthreads();

    // Phase 2: 16 WMMAs over this j-block, both operands via ds_load_b64.
    const float* prow = sP  + idx16 * PSTR + khalf;
    const float* brow = sGT + (wave * 16 + idx16) * PSTR + khalf;
#pragma unroll 4
    for (int kk = 0; kk < 64; kk += 4) {
      v2f a = *(const v2f*)(prow + kk);
      v2f b = *(const v2f*)(brow + kk);
      acc = wmma_f32_k4(a, b, acc);
    }
    __syncthreads();   // protect sP/sGT before next phase-1 overwrite
  }

  const int row = i0 + (lane >> 4) * 8;
  float* orow = out + (size_t)row * OUT_F + head * HIDDEN + wave * 16 + idx16;
#pragma unroll
  for (int v = 0; v < 8; ++v) orow[(size_t)v * OUT_F] = acc[v];
}

// ---------------------------------------------------------------------------
extern "C" void kernel_launch(void* const* d_in, const int* in_sizes, int n_in,
                              void* d_out, int out_size, void* d_ws, size_t ws_size,
                              hipStream_t stream) {
  const float*         h      = (const float*)d_in[0];
  const unsigned char* adj    = (const unsigned char*)d_in[1];  // jax bool = 1 byte
  const float*         W      = (const float*)d_in[2];
  const float*         attn_w = (const float*)d_in[3];
  float*               out    = (float*)d_out;

  // Workspace layout (~4.25 MB total)
  char*  ws  = (char*)d_ws;
  float* g   = (float*)(ws);                              // 4096*256 f32 = 4 MB
  float* sl  = (float*)(ws + (size_t)4194304);            // 4096*4
  float* sr  = (float*)(ws + (size_t)4194304 + 65536);    // 4096*4
  float* Mst = (float*)(ws + (size_t)4194304 + 131072);   // 4096*4
  float* IZ  = (float*)(ws + (size_t)4194304 + 196608);   // 4096*4

  gemm_g_kernel<<<dim3(N_NODES / 16, OUT_F / 64), 128, 0, stream>>>(h, W, g);
  scores_kernel<<<dim3((N_NODES * HEADS) / 256), 256, 0, stream>>>(g, attn_w, sl, sr);
  softmax_stats_kernel<<<dim3(N_NODES), 256, 0, stream>>>(sl, sr, adj, Mst, IZ);
  attn_gemm_kernel<<<dim3(N_NODES / 16, HEADS), 128, 0, stream>>>(g, sl, sr, Mst, IZ, adj, out);
}